// ModularFanBeamCTProjector_89610197664369
// MI455X (gfx1250) — compile-verified
//
#include <hip/hip_runtime.h>

typedef float v2f __attribute__((ext_vector_type(2)));
typedef float v8f __attribute__((ext_vector_type(8)));

// ---- geometry constants (from reference init_kwargs) ----
constexpr int   R = 384, C = 384;
constexpr float DR = 1.0f, DC = 1.0f;
constexpr int   NCOLS = 768;
constexpr float DSPACE = 1.5f;
constexpr float INV_DSPACE = 1.0f / DSPACE;
constexpr int   V = 256;

// ---- tiling ----
constexpr int VG = 16;                       // views per block (WMMA M dim)
constexpr int NSLAB = 64;                    // voxel slabs per view-group
constexpr int NVOX = R * C;                  // 147456
constexpr int VOX_PER_SLAB = NVOX / NSLAB;   // 2304 (divisible by 16 and by C)
constexpr int PITCH = NCOLS + 4;             // 772: rows 1 apart -> +4 banks, rows 8 apart -> +32 banks
constexpr int THREADS = 256;                 // 8 wave32

__global__ __launch_bounds__(THREADS)
void fanbeam_fwd_wmma(const float* __restrict__ img,
                      const float* __restrict__ srcp,   // (V,1,2)
                      const float* __restrict__ detc,   // (V,1,2)
                      const float* __restrict__ uvec,   // (V,1,2)
                      const float* __restrict__ icen,   // (2)
                      const float* __restrict__ cdir,   // (2)
                      float* __restrict__ out)          // (V,1,1,NCOLS)
{
  __shared__ float s_nx[VG], s_ny[VG], s_nsd[VG];   // n, -src·n
  __shared__ float s_ux[VG], s_uy[VG], s_nsu[VG];   // u, -src·u
  __shared__ float s_num[VG], s_ac[VG];             // ref_d - src_d, src_u - zero_u
  __shared__ float s_sino[VG * PITCH];              // per-block private sinogram tile

  const int tid      = threadIdx.x;
  const int vg       = blockIdx.x / NSLAB;
  const int slab     = blockIdx.x % NSLAB;
  const int viewBase = vg * VG;

  // ---- per-view geometry (threads 0..15), exactly as reference ----
  if (tid < VG) {
    int v = viewBase + tid;
    float sx = srcp[2 * v], sy = srcp[2 * v + 1];
    float dx = detc[2 * v], dy = detc[2 * v + 1];
    float ux = uvec[2 * v], uy = uvec[2 * v + 1];
    float nx = -uy, ny = ux;                       // detector normal
    float ref_d = dx * nx + dy * ny;
    float src_d = sx * nx + sy * ny;
    float src_u = sx * ux + sy * uy;
    float zx = dx - 0.5f * (NCOLS - 1) * DSPACE * ux;   // det_zero
    float zy = dy - 0.5f * (NCOLS - 1) * DSPACE * uy;
    float zero_u = zx * ux + zy * uy;
    s_nx[tid] = nx;  s_ny[tid] = ny;  s_nsd[tid] = -src_d;
    s_ux[tid] = ux;  s_uy[tid] = uy;  s_nsu[tid] = -src_u;
    s_num[tid] = ref_d - src_d;
    s_ac[tid]  = src_u - zero_u;
  }
  // zero the private LDS sinogram
  for (int i = tid; i < VG * PITCH; i += THREADS) s_sino[i] = 0.0f;
  __syncthreads();

  // ---- image-frame constants (uniform scalars) ----
  float cdx = cdir[0], cdy = cdir[1];
  float rdx = -cdy, rdy = cdx;                     // row_dir
  float vzx = icen[0] - 0.5f * (R - 1) * DR * rdx - 0.5f * (C - 1) * DC * cdx;
  float vzy = icen[1] - 0.5f * (R - 1) * DR * rdy - 0.5f * (C - 1) * DC * cdy;

  const int  lane = tid & 31;        // wave32
  const int  lv   = lane & 15;
  const bool hi   = lane >= 16;

  // ---- A matrices (16 views x K4), per ISA 32-bit A layout:
  // VGPR0: lanes0-15 = K0, lanes16-31 = K2 ; VGPR1: K1 / K3
  v2f a_den, a_u;
  a_den.x = hi ? s_nsd[lv] : s_nx[lv];   // K2 = -src·n  | K0 = n.x
  a_den.y = hi ? 0.0f      : s_ny[lv];   // K3 = 0       | K1 = n.y
  a_u.x   = hi ? s_nsu[lv] : s_ux[lv];
  a_u.y   = hi ? 0.0f      : s_uy[lv];

  // loop-invariant per-D-component view constants (D: VGPR r -> view r or r+8)
  float cnum[8], cac[8];
  int   rowoff[8];
#pragma unroll
  for (int r = 0; r < 8; ++r) {
    int vv = r + (hi ? 8 : 0);
    cnum[r]   = s_num[vv];
    cac[r]    = s_ac[vv];
    rowoff[r] = vv * PITCH;
  }

  const int slabStart = slab * VOX_PER_SLAB;
  const int nTiles    = VOX_PER_SLAB / 16;   // 144, divisible by 8 waves
  const int wave      = tid >> 5;

  for (int tile = wave; tile < nTiles; tile += THREADS / 32) {
    int voxel = slabStart + tile * 16 + lv;
    int rr = voxel / C;
    int cc = voxel - rr * C;
    float px = vzx + rr * (DR * rdx) + cc * (DC * cdx);
    float py = vzy + rr * (DR * rdy) + cc * (DC * cdy);

    // B matrix (K4 x 16 voxels): VGPR0: K0 row | K2 row ; VGPR1: K1 | K3
    v2f b;
    b.x = hi ? 1.0f : px;
    b.y = hi ? 0.0f : py;

    float imgv = img[voxel];                 // lanes 16-31 re-read same 16 values (cache hit)

    v8f zero8 = {};
    v8f den = __builtin_amdgcn_wmma_f32_16x16x4_f32(false, a_den, false, b,
                                                    (short)0, zero8, false, false);
    v8f q   = __builtin_amdgcn_wmma_f32_16x16x4_f32(false, a_u,   false, b,
                                                    (short)0, zero8, false, false);

#pragma unroll
    for (int r = 0; r < 8; ++r) {
      // fast reciprocal (v_rcp_f32, ~1 ulp): splat is continuous in colf,
      // so ulp-level error here perturbs the output only at ulp scale.
      float t = cnum[r] * __builtin_amdgcn_rcpf(den[r]);
      float colf = (cac[r] + t * q[r]) * INV_DSPACE;
      float c0 = floorf(colf);
      float w1 = colf - c0;
      int   i0 = (int)c0;                    // floor value, exact conversion
      bool fwd = t > 0.0f;
      bool m0  = fwd && (i0 >= 0) && (i0 < NCOLS);
      bool m1  = fwd && (i0 >= -1) && (i0 < NCOLS - 1);
      int  j0  = min(max(i0, 0), NCOLS - 1);
      int  j1  = min(max(i0 + 1, 0), NCOLS - 1);
      // branchless masked splat: matches reference (adds +/-0 when masked)
      atomicAdd(&s_sino[rowoff[r] + j0], m0 ? (1.0f - w1) * imgv : 0.0f);  // ds_add_f32
      atomicAdd(&s_sino[rowoff[r] + j1], m1 ? w1 * imgv : 0.0f);           // ds_add_f32
    }
  }
  __syncthreads();

  // ---- flush private sinogram tile to global with float atomics ----
  for (int i = tid; i < VG * NCOLS; i += THREADS) {
    int v = i / NCOLS;
    int c = i - v * NCOLS;
    float val = s_sino[v * PITCH + c];
    unsafeAtomicAdd(&out[(viewBase + v) * NCOLS + c], val);   // global_atomic_add_f32
  }
}

extern "C" void kernel_launch(void* const* d_in, const int* in_sizes, int n_in,
                              void* d_out, int out_size, void* d_ws, size_t ws_size,
                              hipStream_t stream) {
  const float* img  = (const float*)d_in[0];
  const float* srcp = (const float*)d_in[1];
  const float* detc = (const float*)d_in[2];
  const float* uvec = (const float*)d_in[3];
  const float* icen = (const float*)d_in[4];
  const float* cdir = (const float*)d_in[5];
  float* out = (float*)d_out;

  // output is accumulated into -> must start at zero every launch (capture-safe)
  hipMemsetAsync(out, 0, sizeof(float) * (size_t)out_size, stream);

  dim3 grid((V / VG) * NSLAB);   // 16 view-groups x 64 voxel slabs = 1024 blocks
  fanbeam_fwd_wmma<<<grid, THREADS, 0, stream>>>(img, srcp, detc, uvec, icen, cdir, out);
}